// PeakDetector_77438260347440
// MI455X (gfx1250) — compile-verified
//
#include <hip/hip_runtime.h>
#include <stdint.h>

// Problem constants (from reference)
#define BATCH   16
#define VOCAB   100000
#define DIM     64
#define TOPK    64

// Tiling
#define CHUNK   1024                           // vocab rows per block
#define NCHUNK  ((VOCAB + CHUNK - 1) / CHUNK)  // 98
#define NWAVE   8
#define ROWS_PER_PASS (NWAVE * 16)             // 128
#define NPASS   (CHUNK / ROWS_PER_PASS)        // 8
#define RSTRIDE 66                             // padded LDS row stride (floats), 8B aligned
#define NCAND   (NCHUNK * TOPK)                // 6272 candidates per batch
#define MSEG    (NCAND / 32)                   // 196 merge entries per lane

typedef float v2f __attribute__((ext_vector_type(2)));
typedef float v8f __attribute__((ext_vector_type(8)));
typedef unsigned int u32x4 __attribute__((ext_vector_type(4)));
typedef int i32x4 __attribute__((ext_vector_type(4)));
typedef int i32x8 __attribute__((ext_vector_type(8)));
typedef unsigned long long u64;

#if __has_builtin(__builtin_amdgcn_wmma_f32_16x16x4_f32)
#define HAVE_WMMA_F32 1
#else
#define HAVE_WMMA_F32 0
#endif

#if __has_builtin(__builtin_amdgcn_tensor_load_to_lds)
#define HAVE_TDM 1
#else
#define HAVE_TDM 0
#endif

// Monotone float->uint mapping: a > b (float)  <=>  ordf(a) > ordf(b)
__device__ __forceinline__ unsigned ordf(float f) {
  unsigned u = __float_as_uint(f);
  return (u & 0x80000000u) ? ~u : (u | 0x80000000u);
}

// ---------------------------------------------------------------------------
// Kernel 1: per-(batch, chunk) scoring via WMMA + per-block exact top-64.
// Scores never touch HBM; only 64 candidate keys per chunk are written.
// Bias is a constant shift -> irrelevant to top-k ordering -> dropped.
// ---------------------------------------------------------------------------
__global__ __launch_bounds__(256) void pd_score_topk(
    const float* __restrict__ act, const float* __restrict__ w,
    u64* __restrict__ cand)
{
  __shared__ float tile[NWAVE * 16 * RSTRIDE];   // per-wave 16x64 tiles (padded)
  __shared__ float wbuf[DIM];
  __shared__ float scoreTile[CHUNK];
  __shared__ u64   keyTile[CHUNK];

  const int tid  = threadIdx.x;
  const int wave = tid >> 5;
  const int lane = tid & 31;
  const int lrow = lane & 15;
  const int half = lane >> 4;
  const int b     = blockIdx.y;
  const int chunk = blockIdx.x;
  const int start = chunk * CHUNK;

  if (tid < DIM) wbuf[tid] = w[tid];
  __syncthreads();

  float* mytile = &tile[wave * 16 * RSTRIDE];

#if HAVE_WMMA_F32
  // Hoist B (w broadcast) into 32 VGPRs once: same per-lane K pair as A.
  v2f breg[16];
  #pragma unroll
  for (int i = 0; i < 16; ++i)
    breg[i] = *(const v2f*)(&wbuf[4 * i + half * 2]);
#endif

  for (int pass = 0; pass < NPASS; ++pass) {
    const int rowBase = pass * ROWS_PER_PASS + wave * 16;  // row within chunk
    const long long gRow = (long long)start + rowBase;     // global vocab row
    const long long remainRows = (long long)VOCAB - gRow;
    if (remainRows > 0) {
      // --- stage 16 rows x 64 f32 into LDS (padded stride 66) ---
#if HAVE_TDM
      // WAR: make sure prior-pass LDS reads of mytile finished before TDM rewrites it
      asm volatile("s_wait_dscnt 0" ::: "memory");
      const u64 gaddr = (u64)(uintptr_t)act +
          (((u64)b * VOCAB + (u64)gRow) * DIM) * 4ull;
      long long remElemLL = remainRows * DIM;
      const unsigned td0 = (remElemLL > 0x7FFFFFFFll) ? 0x7FFFFFFFu : (unsigned)remElemLL;
      const unsigned ldsOff = (unsigned)(uintptr_t)mytile;  // flat low 32 bits = LDS offset

      u32x4 g0;
      g0.x = 1u;                                             // count=1, user descriptor
      g0.y = ldsOff;                                         // lds_addr
      g0.z = (unsigned)(gaddr & 0xFFFFFFFFull);              // global_addr[31:0]
      g0.w = (unsigned)((gaddr >> 32) & 0x1FFFFFFull) | (2u << 30);  // addr[56:32] | type=2

      i32x8 g1;
      g1[0] = (int)((2u << 16)      // data_size = 4B
                  | (1u << 20)      // pad_enable
                  | (5u << 22)      // pad_interval: 64 DWORDs (one 64-f32 row)
                  | (1u << 25));    // pad_amount: 2 DWORDs -> 66-float stride
      g1[1] = (int)((td0 & 0xFFFFu) << 16);                  // tensor_dim0[15:0]
      g1[2] = (int)(((td0 >> 16) & 0xFFFFu) | (1u << 16));   // tensor_dim0[31:16] | tensor_dim1=1
      g1[3] = (int)(1024u << 16);                            // tile_dim0 = 16*64 elements (1D)
      g1[4] = 0;                                             // tile_dim1 = tile_dim2 = 0
      g1[5] = (int)DIM;                                      // tensor_dim0_stride (unused, 1D)
      g1[6] = 0;
      g1[7] = 0;
      i32x4 gz4 = {0, 0, 0, 0};
      i32x8 gz8 = {0, 0, 0, 0, 0, 0, 0, 0};
      __builtin_amdgcn_tensor_load_to_lds(g0, g1, gz4, gz4, gz8, 0);
#if __has_builtin(__builtin_amdgcn_s_wait_tensorcnt)
      __builtin_amdgcn_s_wait_tensorcnt(0);
#else
      asm volatile("s_wait_tensorcnt 0" ::: "memory");
#endif
#else   // manual coalesced staging fallback
      for (int e = lane; e < 16 * DIM; e += 32) {
        const int r = e >> 6, d = e & 63;
        float vv = 0.f;
        if (gRow + r < (long long)VOCAB)
          vv = act[((u64)b * VOCAB + (u64)(gRow + r)) * DIM + d];
        mytile[r * RSTRIDE + d] = vv;
      }
#endif

      // Warm L2 for this wave's next-pass tile while we compute this one.
      {
        const long long nRow = gRow + ROWS_PER_PASS;
        if (pass + 1 < NPASS && nRow < (long long)VOCAB) {
          const char* pf = (const char*)act +
              (((u64)b * VOCAB + (u64)nRow) * DIM) * 4ull + (u64)lane * 128u;
          __builtin_prefetch(pf, 0, 1);
        }
      }

      // --- 16 rows x (K=64) dot with w ---
#if HAVE_WMMA_F32
      // A (16x4 f32): lanes 0-15 hold K=+0,+1 ; lanes 16-31 hold K=+2,+3 (ISA layout).
      v8f c = {0.f, 0.f, 0.f, 0.f, 0.f, 0.f, 0.f, 0.f};
      #pragma unroll
      for (int i = 0; i < 16; ++i) {
        const int ko = 4 * i + half * 2;
        v2f a = *(const v2f*)(&mytile[lrow * RSTRIDE + ko]);
        c = __builtin_amdgcn_wmma_f32_16x16x4_f32(
            false, a, false, breg[i], (short)0, c, false, false);
      }
      // C/D layout: VGPR j = row (j + 8*half), identical across N -> lanes with lrow==0 export
      if (lrow == 0) {
        float* dst = &scoreTile[rowBase + half * 8];
        dst[0] = c[0]; dst[1] = c[1]; dst[2] = c[2]; dst[3] = c[3];
        dst[4] = c[4]; dst[5] = c[5]; dst[6] = c[6]; dst[7] = c[7];
      }
#else
      // VALU fallback: lanes 0-15 each own one row
      if (half == 0) {
        float s = 0.f;
        #pragma unroll
        for (int d = 0; d < DIM; ++d) s = fmaf(mytile[lrow * RSTRIDE + d], wbuf[d], s);
        scoreTile[rowBase + lrow] = s;
      }
#endif
    }
  }

  __syncthreads();
  // Build monotone (score desc, index asc) keys; mask rows beyond the vocab.
  for (int r = tid; r < CHUNK; r += 256) {
    const float s = (start + r < VOCAB) ? scoreTile[r] : -__builtin_huge_valf();
    keyTile[r] = ((u64)ordf(s) << 32) | (u64)(0xFFFFFFFFu - (unsigned)r);
  }
  __syncthreads();

  // Barrier-free exact top-64 tournament, wave 0 only.
  // Lane L owns strided entries {L, L+32, ..., L+992} (bank-conflict-free).
  if (wave == 0) {
    u64* candOut = &cand[((u64)b * NCHUNK + chunk) * TOPK];
    u64 localBest = 0;
    #pragma unroll 4
    for (int j = 0; j < CHUNK / 32; ++j) {
      const u64 v = keyTile[j * 32 + lane];
      localBest = localBest > v ? localBest : v;
    }
    for (int k = 0; k < TOPK; ++k) {
      u64 m = localBest;
      #pragma unroll
      for (int off = 16; off > 0; off >>= 1) {
        const u64 o = __shfl_xor(m, off, 32);
        m = m > o ? m : o;
      }
      if (localBest == m) {            // unique owner: kill winner, rescan segment
        u64 nb = 0;
        #pragma unroll 4
        for (int j = 0; j < CHUNK / 32; ++j) {
          u64 v = keyTile[j * 32 + lane];
          if (v == m) { keyTile[j * 32 + lane] = 0; v = 0; }
          nb = nb > v ? nb : v;
        }
        localBest = nb;
      }
      if (lane == 0) {
        const unsigned r = 0xFFFFFFFFu - (unsigned)(m & 0xFFFFFFFFull);
        // re-key with the GLOBAL vocab index for the merge stage
        candOut[k] = (m & 0xFFFFFFFF00000000ull) |
                     (u64)(0xFFFFFFFFu - (unsigned)(start + (int)r));
      }
    }
  }
}

// ---------------------------------------------------------------------------
// Kernel 2: per-batch merge of 98*64 candidates -> exact global top-64,
// then gather the winning activation rows to the output.
// ---------------------------------------------------------------------------
__global__ __launch_bounds__(256) void pd_merge_gather(
    const u64* __restrict__ cand, const float* __restrict__ act,
    float* __restrict__ out)
{
  __shared__ u64 keys[NCAND];     // ~50 KB — fine under the 320 KB WGP LDS
  __shared__ int topIdx[TOPK];

  const int tid  = threadIdx.x;
  const int wave = tid >> 5;
  const int lane = tid & 31;
  const int b    = blockIdx.x;

  for (int i = tid; i < NCAND; i += 256) keys[i] = cand[(u64)b * NCAND + i];
  __syncthreads();

  // Wave-0 tournament over 6272 keys, strided ownership (196 per lane).
  if (wave == 0) {
    u64 localBest = 0;
    for (int j = 0; j < MSEG; ++j) {
      const u64 v = keys[j * 32 + lane];
      localBest = localBest > v ? localBest : v;
    }
    for (int k = 0; k < TOPK; ++k) {
      u64 m = localBest;
      #pragma unroll
      for (int off = 16; off > 0; off >>= 1) {
        const u64 o = __shfl_xor(m, off, 32);
        m = m > o ? m : o;
      }
      if (localBest == m) {            // unique owner: kill winner, rescan segment
        u64 nb = 0;
        for (int j = 0; j < MSEG; ++j) {
          u64 v = keys[j * 32 + lane];
          if (v == m) { keys[j * 32 + lane] = 0; v = 0; }
          nb = nb > v ? nb : v;
        }
        localBest = nb;
      }
      if (lane == 0)
        topIdx[k] = (int)(0xFFFFFFFFu - (unsigned)(m & 0xFFFFFFFFull));
    }
  }
  __syncthreads();

  // Gather: out[b, k, :] = act[b, topIdx[k], :]  (float4-vectorized)
  const float4* actv = (const float4*)act;
  float4*       outv = (float4*)out;
  for (int t = tid; t < TOPK * (DIM / 4); t += 256) {
    const int k = t >> 4, q = t & 15;
    outv[(u64)b * TOPK * (DIM / 4) + (u64)t] =
        actv[((u64)b * VOCAB + (u64)(unsigned)topIdx[k]) * (DIM / 4) + q];
  }
}

// ---------------------------------------------------------------------------
extern "C" void kernel_launch(void* const* d_in, const int* in_sizes, int n_in,
                              void* d_out, int out_size, void* d_ws, size_t ws_size,
                              hipStream_t stream) {
  (void)in_sizes; (void)n_in; (void)out_size; (void)ws_size;
  const float* act = (const float*)d_in[0];
  const float* w   = (const float*)d_in[1];
  // d_in[2] (bias) intentionally unused: a uniform shift of all scores cannot
  // change top-k ordering, and the output is gathered activations, not scores.
  u64* cand = (u64*)d_ws;   // 16 * 98 * 64 * 8 B = 802,816 B of scratch

  dim3 grid1(NCHUNK, BATCH);
  pd_score_topk<<<grid1, 256, 0, stream>>>(act, w, cand);
  pd_merge_gather<<<BATCH, 256, 0, stream>>>(cand, act, (float*)d_out);
}